// cheerla_et_al_criterion_17093969838725
// MI455X (gfx1250) — compile-verified
//
#include <hip/hip_runtime.h>
#include <hip/hip_bf16.h>
#include <math.h>

#define NPAT 4096
#define DIM  256
#define EPSV 1e-8f

typedef __attribute__((ext_vector_type(2))) float v2f;
typedef __attribute__((ext_vector_type(8))) float v8f;

// ---------------------------------------------------------------------------
// Kernel 1: zero the f64 accumulators, compute exp(h)
// acc[0] = sim sum, acc[1] = cox numerator, acc[2] = event count
// ---------------------------------------------------------------------------
__global__ void k_init(const float* __restrict__ h, float* __restrict__ exph,
                       double* __restrict__ acc) {
    int i = blockIdx.x * blockDim.x + threadIdx.x;
    if (i < NPAT) exph[i] = expf(h[i]);
    if (i < 3) acc[i] = 0.0;
}

// ---------------------------------------------------------------------------
// Kernel 2: per-row preprocessing for all 4 modalities.
// One block (256 threads == DIM) per patient row.
//   nrm   = ||x||, den = max(nrm, eps)
//   Xn    = x / den         (written to workspace)
//   diag  = nrm^2 / den^2
//   missf = 1.0 if row is constant (modality missing), else 0.0
// ---------------------------------------------------------------------------
__global__ void k_prep(const float* __restrict__ e0, const float* __restrict__ e1,
                       const float* __restrict__ e2, const float* __restrict__ e3,
                       float* __restrict__ Xn, float* __restrict__ diag,
                       float* __restrict__ missf) {
    __shared__ float s_red[DIM];
    __shared__ float s_first;
    const int row = blockIdx.x;
    const int tid = threadIdx.x;
    const float* ebs[4] = {e0, e1, e2, e3};

    for (int m = 0; m < 4; ++m) {
        float x = ebs[m][(size_t)row * DIM + tid];
        if (tid == 0) s_first = x;
        __syncthreads();
        const float first = s_first;

        // sum of squares
        s_red[tid] = x * x;
        __syncthreads();
        for (int s = DIM / 2; s > 0; s >>= 1) {
            if (tid < s) s_red[tid] += s_red[tid + s];
            __syncthreads();
        }
        const float sumsq = s_red[0];
        __syncthreads();

        // all-equal-to-first (missing modality) via min-reduce
        s_red[tid] = (x == first) ? 1.0f : 0.0f;
        __syncthreads();
        for (int s = DIM / 2; s > 0; s >>= 1) {
            if (tid < s) s_red[tid] = fminf(s_red[tid], s_red[tid + s]);
            __syncthreads();
        }
        const float allEq = s_red[0];
        __syncthreads();

        const float nrm = sqrtf(sumsq);
        const float den = fmaxf(nrm, EPSV);
        Xn[((size_t)m * NPAT + row) * DIM + tid] = x / den;
        if (tid == 0) {
            diag[m * NPAT + row]  = sumsq / (den * den);
            missf[m * NPAT + row] = allEq;
        }
        __syncthreads();
    }
}

// ---------------------------------------------------------------------------
// Kernel 3: Cox negative partial log-likelihood pieces.
// risk(i,j) = (t_i*t_j >= t_i*t_i); denom_i = sum_j risk * exp(h_j)
// acc[1] += e_i * (h_i - log(denom_i));  acc[2] += e_i
// ---------------------------------------------------------------------------
__global__ void k_cox(const float* __restrict__ h, const int* __restrict__ t,
                      const int* __restrict__ e, const float* __restrict__ exph,
                      double* __restrict__ acc) {
    __shared__ int   s_t[NPAT];
    __shared__ float s_eh[NPAT];
    const int tid = threadIdx.x;
    for (int j = tid; j < NPAT; j += blockDim.x) {
        s_t[j]  = t[j];
        s_eh[j] = exph[j];
    }
    __syncthreads();

    const int i   = blockIdx.x * blockDim.x + tid;
    const int ti  = t[i];
    const int ti2 = ti * ti;  // <= 3650^2, fits int32 (matches reference math)
    float denom = 0.0f;
    for (int j = 0; j < NPAT; ++j) {
        denom += (ti * s_t[j] >= ti2) ? s_eh[j] : 0.0f;
    }
    const int   ev   = (e[i] != 0) ? 1 : 0;
    const float term = ev ? (h[i] - logf(denom)) : 0.0f;

    // wave-level reduce, then one f64 atomic per wave
    float rterm = term;
    float rcnt  = (float)ev;
    for (int off = 16; off; off >>= 1) {
        rterm += __shfl_down(rterm, off, 32);
        rcnt  += __shfl_down(rcnt, off, 32);
    }
    if ((threadIdx.x & 31) == 0) {
        atomicAdd(&acc[1], (double)rterm);
        atomicAdd(&acc[2], (double)rcnt);
    }
}

// ---------------------------------------------------------------------------
// Kernel 4: similarity loss via V_WMMA_F32_16X16X4_F32 Gram tiles.
// Each wave owns one 16x16 (i,j) tile and keeps 4 accumulators (one per
// modality). A and B fragments have identical per-lane layouts for a Gram
// product: lane loads float2 at row (base + lane%16), col (k + 2*(lane/16)).
// C/D layout: VGPR v of lane L holds (i = i0 + v + 8*(L/16), j = j0 + L%16).
// ---------------------------------------------------------------------------
__global__ void k_sim(const float* __restrict__ Xn, const float* __restrict__ diag,
                      const float* __restrict__ missf, const int* __restrict__ mptr,
                      double* __restrict__ acc) {
    const int lane  = threadIdx.x & 31;
    const int wave  = threadIdx.x >> 5;
    const int tileJ = blockIdx.x * 8 + wave;   // gridDim.x = 32  -> 256 J tiles
    const int tileI = blockIdx.y;              // gridDim.y = 256 -> 256 I tiles
    const int l15 = lane & 15;
    const int hi  = lane >> 4;
    const int i0  = tileI * 16;
    const int j0  = tileJ * 16;

    v8f accm[4];
    #pragma unroll
    for (int m = 0; m < 4; ++m) {
        const float* pa = Xn + ((size_t)m * NPAT + (i0 + l15)) * DIM + 2 * hi;
        const float* pb = Xn + ((size_t)m * NPAT + (j0 + l15)) * DIM + 2 * hi;
        v8f c = {0.f, 0.f, 0.f, 0.f, 0.f, 0.f, 0.f, 0.f};
        #pragma unroll 8
        for (int k = 0; k < DIM; k += 4) {
            v2f a = *(const v2f*)(pa + k);   // 8B-aligned global_load_b64
            v2f b = *(const v2f*)(pb + k);
            c = __builtin_amdgcn_wmma_f32_16x16x4_f32(
                    false, a, false, b, (short)0, c, false, false);
        }
        accm[m] = c;
    }

    const float margin = (float)(*mptr);
    const int j = j0 + l15;
    float mj[4];
    #pragma unroll
    for (int m = 0; m < 4; ++m) mj[m] = missf[m * NPAT + j];

    float tsum = 0.0f;
    #pragma unroll
    for (int v = 0; v < 8; ++v) {
        const int i = i0 + v + 8 * hi;
        float matched = 0.0f, own = 0.0f;
        #pragma unroll
        for (int m = 0; m < 4; ++m) {
            const float valid =
                ((mj[m] != 0.0f) || (missf[m * NPAT + i] != 0.0f)) ? 0.0f : 1.0f;
            matched += valid * accm[m][v];
            own     += valid * diag[m * NPAT + i];
        }
        float pp = margin - matched + own;
        pp = pp > 0.0f ? pp : 0.0f;
        if (i != j) tsum += pp;
    }
    for (int off = 16; off; off >>= 1) tsum += __shfl_down(tsum, off, 32);
    if (lane == 0) atomicAdd(&acc[0], (double)tsum);
}

// ---------------------------------------------------------------------------
// Kernel 5: combine:  out = -coxnum/esum + sim
// ---------------------------------------------------------------------------
__global__ void k_final(const double* __restrict__ acc, float* __restrict__ out) {
    const double cox = -acc[1] / acc[2];
    out[0] = (float)(cox + acc[0]);
}

// ---------------------------------------------------------------------------
extern "C" void kernel_launch(void* const* d_in, const int* in_sizes, int n_in,
                              void* d_out, int out_size, void* d_ws, size_t ws_size,
                              hipStream_t stream) {
    (void)in_sizes; (void)n_in; (void)out_size; (void)ws_size;
    const float* h   = (const float*)d_in[0];
    const int*   t   = (const int*)d_in[1];
    const int*   e   = (const int*)d_in[2];
    const float* eb0 = (const float*)d_in[3];
    const float* eb1 = (const float*)d_in[4];
    const float* eb2 = (const float*)d_in[5];
    const float* eb3 = (const float*)d_in[6];
    const int*   mrg = (const int*)d_in[7];

    float* Xn    = (float*)d_ws;                       // 4*N*D floats (16 MB)
    float* diag  = Xn + (size_t)4 * NPAT * DIM;        // 4*N
    float* missf = diag + 4 * NPAT;                    // 4*N
    float* exph  = missf + 4 * NPAT;                   // N
    double* acc  = (double*)(exph + NPAT);             // 3 doubles (8B aligned)

    float* out = (float*)d_out;

    k_init<<<16, 256, 0, stream>>>(h, exph, acc);
    k_prep<<<NPAT, DIM, 0, stream>>>(eb0, eb1, eb2, eb3, Xn, diag, missf);
    k_cox<<<16, 256, 0, stream>>>(h, t, e, exph, acc);
    k_sim<<<dim3(32, 256), 256, 0, stream>>>(Xn, diag, missf, mrg, acc);
    k_final<<<1, 1, 0, stream>>>(acc, out);
}